// SelfWarp_74320114090142
// MI455X (gfx1250) — compile-verified
//
#include <hip/hip_runtime.h>

typedef float v2f __attribute__((ext_vector_type(2)));
typedef float v4f __attribute__((ext_vector_type(4)));
typedef float v8f __attribute__((ext_vector_type(8)));

#define HH     128
#define WW     128
#define CIN    256
#define COUT   256
#define NHEADS 32
#define HW     (HH * WW)
#define LDSW   136   // padded row stride (floats): kills 2-way LDS bank conflicts

// Fused dual-output GEMM panel: per K-step, 8 A fragments (LDS) are loaded ONCE
// and feed 16 WMMAs (8 accumulators for each of two weight matrices).
__device__ inline void gemm_panel_dual8(const float* __restrict__ lds_a,
                                        const float* __restrict__ Wa,
                                        const float* __restrict__ Wb,
                                        int n_base, int K, int lane,
                                        v8f* acca, v8f* accb)
{
    const int nlo   = lane & 15;
    const int khalf = (lane >> 4) * 2;           // lanes 0-15: k0,k1 ; 16-31: k2,k3
    const float* warow = Wa + (size_t)(n_base + nlo) * K + khalf;
    const float* wbrow = Wb + (size_t)(n_base + nlo) * K + khalf;
    __builtin_prefetch(warow, 0, 1);             // global_prefetch_b8
    __builtin_prefetch(wbrow, 0, 1);
#pragma unroll 1
    for (int k = 0; k < K; k += 4) {
        v2f bfa = *(const v2f*)(warow + k);
        v2f bfb = *(const v2f*)(wbrow + k);
        const float* arow = lds_a + (k + khalf) * LDSW + nlo;
        v2f a[8];
#pragma unroll
        for (int t = 0; t < 8; ++t) {
            a[t].x = arow[t * 16];
            a[t].y = arow[t * 16 + LDSW];
        }
#pragma unroll
        for (int t = 0; t < 8; ++t)
            acca[t] = __builtin_amdgcn_wmma_f32_16x16x4_f32(
                false, a[t], false, bfa, (short)0, acca[t], false, false);
#pragma unroll
        for (int t = 0; t < 8; ++t)
            accb[t] = __builtin_amdgcn_wmma_f32_16x16x4_f32(
                false, a[t], false, bfb, (short)0, accb[t], false, false);
    }
}

// Single-output panel (pass 2): NMT independent accumulators share one B frag.
template <int NMT>
__device__ inline void gemm_panel(const float* __restrict__ lds_a, int mt0,
                                  const float* __restrict__ Wmat, int n_base,
                                  int K, int lane, v8f* acc)
{
    const int nlo   = lane & 15;
    const int khalf = (lane >> 4) * 2;
    const float* wrow = Wmat + (size_t)(n_base + nlo) * K + khalf;
    __builtin_prefetch(wrow, 0, 1);
#pragma unroll 2
    for (int k = 0; k < K; k += 4) {
        v2f bfrag = *(const v2f*)(wrow + k);
        const float* arow = lds_a + (k + khalf) * LDSW + mt0 * 16 + nlo;
#pragma unroll
        for (int t = 0; t < NMT; ++t) {
            v2f a;
            a.x = arow[t * 16];
            a.y = arow[t * 16 + LDSW];
            acc[t] = __builtin_amdgcn_wmma_f32_16x16x4_f32(
                false, a, false, bfrag, (short)0, acc[t], false, false);
        }
    }
}

// One workgroup (256 threads = 8 waves) per image row (b, y): 128 pixels.
// Pass 1 (fused): h = relu(u@w1^T+b1) -> LDS  AND  value = u@wv^T+bv -> ws,
//   sharing A fragments; wave w owns N-tiles 2w, 2w+1 across all 8 M-tiles.
// Pass 2: flow = h@w2^T+b2 -> ws; wave w owns N-tile w>>1, 4 M-tiles.
__global__ void __launch_bounds__(256)
fused_gemm_kernel(const float* __restrict__ u,
                  const float* __restrict__ w1, const float* __restrict__ b1,
                  const float* __restrict__ w2, const float* __restrict__ b2,
                  const float* __restrict__ wv, const float* __restrict__ bv,
                  float* __restrict__ value_ws, float* __restrict__ flow_ws)
{
    extern __shared__ float smem[];
    float* lds_u = smem;                  // [CIN][LDSW]
    float* lds_h = smem + CIN * LDSW;     // [COUT][LDSW]

    const int tid  = threadIdx.x;
    const int lane = tid & 31;
    const int wave = tid >> 5;
    const int wg   = blockIdx.x;
    const int b    = wg >> 7;
    const int y    = wg & 127;

    // ---- async-stage u[b, :, y, :] (256 ch x 128 px) into LDS [c][x] ----
    {
        const float* src = u + ((size_t)b * CIN * HH + y) * WW;  // block-uniform
        const unsigned x4 = (tid & 31) * 4;
        const unsigned c0 = tid >> 5;
        for (int it = 0; it < 32; ++it) {
            unsigned c = it * 8 + c0;
            unsigned goff = (c * HW + x4) * 4u;                       // byte offset
            unsigned lds_off = (unsigned)(size_t)(lds_u + c * LDSW + x4);
            asm volatile("global_load_async_to_lds_b128 %0, %1, %2"
                         :: "v"(lds_off), "v"(goff), "s"(src)
                         : "memory");
        }
        asm volatile("s_wait_asynccnt 0x0" ::: "memory");
    }
    __syncthreads();

    const int nlo  = lane & 15;
    const int mofs = (lane >> 4) * 8;     // m = mt*16 + mofs + r for acc reg r

    // ---- pass 1 (fused h + value) ----
    for (int j = 0; j < 2; ++j) {
        const int n_base = (wave * 2 + j) * 16;
        const int n = n_base + nlo;
        const v8f z = {};

        v8f acch[8], accv[8];
#pragma unroll
        for (int t = 0; t < 8; ++t) { acch[t] = z; accv[t] = z; }
        gemm_panel_dual8(lds_u, w1, wv, n_base, CIN, lane, acch, accv);

        // h epilogue: bias + relu -> LDS [n][m]
        const float bias1 = b1[n];
#pragma unroll
        for (int t = 0; t < 8; ++t) {
            float* hd = lds_h + n * LDSW + t * 16 + mofs;
            v4f s0, s1;
#pragma unroll
            for (int r = 0; r < 4; ++r) {
                float v0 = acch[t][r] + bias1;
                float v1 = acch[t][r + 4] + bias1;
                s0[r] = v0 > 0.0f ? v0 : 0.0f;
                s1[r] = v1 > 0.0f ? v1 : 0.0f;
            }
            *(v4f*)(hd)     = s0;
            *(v4f*)(hd + 4) = s1;
        }

        // value epilogue: bias -> global workspace (coalesced along x)
        const float biasv = bv[n];
        float* vrow = value_ws + ((size_t)(b * COUT + n) * HH + y) * WW + mofs;
#pragma unroll
        for (int t = 0; t < 8; ++t) {
            v4f s0, s1;
#pragma unroll
            for (int r = 0; r < 4; ++r) {
                s0[r] = accv[t][r] + biasv;
                s1[r] = accv[t][r + 4] + biasv;
            }
            *(v4f*)(vrow + t * 16)     = s0;
            *(v4f*)(vrow + t * 16 + 4) = s1;
        }
    }
    __syncthreads();

    // ---- pass 2: flow = h @ w2^T + b2 (N = 64) ----
    {
        const int n_base = (wave >> 1) * 16;
        const int mt0    = (wave & 1) * 4;
        const int n      = n_base + nlo;
        const v8f z = {};
        v8f acc[4];
#pragma unroll
        for (int t = 0; t < 4; ++t) acc[t] = z;
        gemm_panel<4>(lds_h, mt0, w2, n_base, COUT, lane, acc);
        const float bias2 = b2[n];
        const int head = n >> 1, d = n & 1;
        float* frow = flow_ws + (((size_t)(b * NHEADS + head) * HH + y) * WW) * 2 + d;
#pragma unroll
        for (int t = 0; t < 4; ++t) {
            const int m0 = (mt0 + t) * 16 + mofs;
#pragma unroll
            for (int r = 0; r < 8; ++r)
                frow[(m0 + r) * 2] = acc[t][r] + bias2;
        }
    }
}

// One thread per (b, head, y, x); bilinear gather of 8 channels, zeros padding.
__global__ void __launch_bounds__(256)
warp_kernel(const float* __restrict__ value_ws, const float* __restrict__ flow_ws,
            float* __restrict__ out)
{
    const int idx  = blockIdx.x * 256 + threadIdx.x;
    const int x    = idx & 127;
    const int y    = (idx >> 7) & 127;
    const int head = (idx >> 14) & 31;
    const int b    = idx >> 19;

    const float fx = flow_ws[(size_t)idx * 2 + 0];
    const float fy = flow_ws[(size_t)idx * 2 + 1];
    const float ix = (float)x + fx * 63.5f;   // (gx+1)*(W-1)/2 with gx = base+fx
    const float iy = (float)y + fy * 63.5f;
    const float x0f = floorf(ix), y0f = floorf(iy);
    const float dx = ix - x0f, dy = iy - y0f;
    const int x0 = (int)x0f, y0 = (int)y0f;

    float acc[8];
#pragma unroll
    for (int c = 0; c < 8; ++c) acc[c] = 0.0f;

    const float* vbase = value_ws + (size_t)(b * COUT + head * 8) * HW;
    const float wgt[4] = { (1.0f - dx) * (1.0f - dy), dx * (1.0f - dy),
                           (1.0f - dx) * dy,          dx * dy };
#pragma unroll
    for (int corner = 0; corner < 4; ++corner) {
        const int xi = x0 + (corner & 1);
        const int yi = y0 + (corner >> 1);
        if (xi >= 0 && xi < WW && yi >= 0 && yi < HH) {
            const float* vp = vbase + yi * WW + xi;
            const float wc = wgt[corner];
#pragma unroll
            for (int c = 0; c < 8; ++c) acc[c] += wc * vp[(size_t)c * HW];
        }
    }
    float* op = out + (size_t)(b * COUT + head * 8) * HW + y * WW + x;
#pragma unroll
    for (int c = 0; c < 8; ++c) op[(size_t)c * HW] = acc[c];
}

extern "C" void kernel_launch(void* const* d_in, const int* in_sizes, int n_in,
                              void* d_out, int out_size, void* d_ws, size_t ws_size,
                              hipStream_t stream)
{
    const float* u  = (const float*)d_in[0];
    const float* w1 = (const float*)d_in[1];
    const float* b1 = (const float*)d_in[2];
    const float* w2 = (const float*)d_in[3];
    const float* b2 = (const float*)d_in[4];
    const float* wv = (const float*)d_in[5];
    const float* bv = (const float*)d_in[6];

    float* value_ws = (float*)d_ws;                          // 4*256*128*128 f32 = 64 MB
    float* flow_ws  = value_ws + (size_t)4 * COUT * HW;      // 4*32*128*128*2 f32 = 16 MB
    float* out      = (float*)d_out;

    const size_t lds_bytes = (size_t)(CIN + COUT) * LDSW * sizeof(float); // 272 KB of 320 KB
    fused_gemm_kernel<<<dim3(4 * HH), dim3(256), lds_bytes, stream>>>(
        u, w1, b1, w2, b2, wv, bv, value_ws, flow_ws);

    const int total = 4 * NHEADS * HH * WW;                  // 2,097,152 threads
    warp_kernel<<<dim3(total / 256), dim3(256), 0, stream>>>(value_ws, flow_ws, out);
}